// FrequencyDomainIPNLMS_8924942041338
// MI455X (gfx1250) — compile-verified
//
#include <hip/hip_runtime.h>
#include <stdint.h>

// Problem constants (match reference)
#define BB    8
#define TT    2048
#define FF    257
#define NTAP  8
#define STEPS (BB * TT)          // 16384 sequential scan steps
#define GRP   8                  // steps per group (aligned with b-cycle: s=8g+p -> t=g, b=p)
#define NGRP  (STEPS / GRP)      // 2048 groups

#define MU_C     0.5f
#define ALPHA_C  0.5f
#define BETA_C   1e-8f
#define EPS_C    1e-10f
#define MAXUPD   0.01f
#define MAXCOEF  2.0f
#define KL_BASE  ((1.0f - ALPHA_C) / (2.0f * (float)NTAP))   // 0.03125

typedef __attribute__((address_space(1))) int gm_i32;    // global (device) addrspace
typedef __attribute__((address_space(3))) int lds_i32;   // LDS addrspace

__device__ __forceinline__ void async_g2l_b32(const float* g, float* l) {
#if __has_builtin(__builtin_amdgcn_global_load_async_to_lds_b32)
  __builtin_amdgcn_global_load_async_to_lds_b32(
      (gm_i32*)(uintptr_t)g,
      (lds_i32*)(uint32_t)(uintptr_t)l,   // low 32 bits of generic LDS ptr = LDS offset
      0, 0);
#else
  uint32_t lo = (uint32_t)(uintptr_t)l;
  asm volatile("global_load_async_to_lds_b32 %0, %1, off"
               :: "v"(lo), "v"(g) : "memory");
#endif
}

#if __has_builtin(__builtin_amdgcn_s_wait_asynccnt)
#define WAIT_ASYNC(n) do { __builtin_amdgcn_s_wait_asynccnt(n); asm volatile("" ::: "memory"); } while (0)
#else
#define WAIT_ASYNC(n) asm volatile("s_wait_asynccnt %0" :: "i"(n) : "memory")
#endif

__device__ __forceinline__ float fast_rcp(float x) {
#if defined(__HIP_DEVICE_COMPILE__)
  return __builtin_amdgcn_rcpf(x);
#else
  return 1.0f / x;
#endif
}
__device__ __forceinline__ float fast_rsq(float x) {
#if defined(__HIP_DEVICE_COMPILE__)
  return __builtin_amdgcn_rsqf(x);
#else
  return 1.0f / sqrtf(x);
#endif
}
__device__ __forceinline__ float clamp_upd(float x) {
#if defined(__HIP_DEVICE_COMPILE__)
  return __builtin_amdgcn_fmed3f(x, -MAXUPD, MAXUPD);   // v_med3_f32
#else
  return fminf(fmaxf(x, -MAXUPD), MAXUPD);
#endif
}

__global__ __launch_bounds__(32) void aec_ipnlms_kernel(
    const float* __restrict__ mic_r, const float* __restrict__ mic_i,
    const float* __restrict__ ref_r, const float* __restrict__ ref_i,
    const float* __restrict__ c0r,  const float* __restrict__ c0i,
    float* __restrict__ out)
{
  // Double-buffered staging: [parity][p][lane][{mr,mi,rr,ri}] = 8 KB
  __shared__ float sbuf[2][GRP][32][4];

  const int lane = (int)threadIdx.x;
  const int f    = (int)blockIdx.x * 32 + lane;
  // Lanes with f >= FF clamp to FF-1: they follow a bit-identical trajectory to
  // lane f=FF-1 (same inputs, same init), so their stores are harmless
  // duplicates -> no predication needed anywhere.
  const int fc   = (f < FF) ? f : (FF - 1);

  // Per-lane filter state, entirely in VGPRs.
  float cr[NTAP], ci[NTAP], hr[NTAP], hi[NTAP];
#pragma unroll
  for (int n = 0; n < NTAP; ++n) {
    cr[n] = c0r[fc * NTAP + n];
    ci[n] = c0i[fc * NTAP + n];
    hr[n] = 0.0f;
    hi[n] = 0.0f;
  }

  const int plane = BB * TT * FF;
  float* __restrict__ err_r = out;
  float* __restrict__ err_i = out + plane;
  float* __restrict__ ech_r = out + 2 * plane;
  float* __restrict__ ech_i = out + 3 * plane;

  // Issue one group's 32 async loads (4 arrays x 8 steps, 4B/lane each).
  auto issue_group = [&](int par, int base) {
#pragma unroll
    for (int p = 0; p < GRP; ++p) {
      float* slot = &sbuf[par][p][lane][0];
      const int gi = base + p * (TT * FF);
      async_g2l_b32(mic_r + gi, slot + 0);
      async_g2l_b32(mic_i + gi, slot + 1);
      async_g2l_b32(ref_r + gi, slot + 2);
      async_g2l_b32(ref_i + gi, slot + 3);
    }
  };

  // One adaptive-filter step; history "shift" is renamed away by unrolling.
  auto step = [&](const float4 v4, int idx) {
    const float mr = v4.x, mi = v4.y, rr = v4.z, ri = v4.w;

    // push new ref frame into history (eliminated by SROA across unrolled steps)
#pragma unroll
    for (int n = NTAP - 1; n > 0; --n) { hr[n] = hr[n - 1]; hi[n] = hi[n - 1]; }
    hr[0] = rr; hi[0] = ri;

    // echo = sum conj(coef)*hist ; trp = ||hist||^2   (two partial chains each)
    float er0 = 0.f, er1 = 0.f, ei0 = 0.f, ei1 = 0.f, tp0 = 0.f, tp1 = 0.f;
#pragma unroll
    for (int n = 0; n < NTAP; n += 2) {
      er0 = fmaf(cr[n],     hr[n],     fmaf(ci[n],     hi[n],     er0));
      er1 = fmaf(cr[n + 1], hr[n + 1], fmaf(ci[n + 1], hi[n + 1], er1));
      ei0 = fmaf(cr[n],     hi[n],     fmaf(-ci[n],     hr[n],     ei0));
      ei1 = fmaf(cr[n + 1], hi[n + 1], fmaf(-ci[n + 1], hr[n + 1], ei1));
      tp0 = fmaf(hr[n],     hr[n],     fmaf(hi[n],     hi[n],     tp0));
      tp1 = fmaf(hr[n + 1], hr[n + 1], fmaf(hi[n + 1], hi[n + 1], tp1));
    }
    const float echo_r = er0 + er1;
    const float echo_i = ei0 + ei1;
    const float trp    = tp0 + tp1;
    const float er = mr - echo_r;
    const float ei = mi - echo_i;

    // IPNLMS proportional gains
    float cmag2[NTAP];
#pragma unroll
    for (int n = 0; n < NTAP; ++n)
      cmag2[n] = fmaf(cr[n], cr[n], ci[n] * ci[n]);
    // pairwise-tree sum (short latency chain); reference adds EPS twice
    const float ssum = (((cmag2[0] + cmag2[1]) + (cmag2[2] + cmag2[3])) +
                        ((cmag2[4] + cmag2[5]) + (cmag2[6] + cmag2[7]))) +
                       2.0f * EPS_C;
    const float kscale = (1.0f + ALPHA_C) * fast_rcp(ssum);
    const float mun    = MU_C * fast_rcp(trp + (BETA_C + EPS_C));

#pragma unroll
    for (int n = 0; n < NTAP; ++n) {
      const float kl = fmaf(kscale, cmag2[n], KL_BASE);
      const float mk = mun * kl;
      const float pr = fmaf(hr[n], er, hi[n] * ei);
      const float pi = fmaf(hi[n], er, -hr[n] * ei);
      const float ur = clamp_upd(mk * pr);
      const float ui = clamp_upd(mk * pi);
      const float nr = cr[n] + ur;
      const float ni = ci[n] + ui;
      // magnitude cap: min(1, MAXCOEF/sqrt(m2)) == where(cmag>2, 2/cmag, 1)
      const float m2 = fmaf(nr, nr, fmaf(ni, ni, 1e-10f));
      const float sc = fminf(1.0f, MAXCOEF * fast_rsq(m2));
      cr[n] = nr * sc;
      ci[n] = ni * sc;
    }

    // Unconditional stores (inactive lanes duplicate f=FF-1's identical values)
    err_r[idx] = er;
    err_i[idx] = ei;
    ech_r[idx] = echo_r;
    ech_i[idx] = echo_i;
  };

  // ---- pipelined group loop ----
  // step s = 8g + p  ->  t = g, b = p  ->  idx = p*(T*F) + g*F + f
  issue_group(0, fc);            // group 0 -> parity 0
  int gbase = fc;                // = fc + g*FF
  for (int g = 0; g < NGRP; ++g) {
    if (g + 1 < NGRP) {
      issue_group((g & 1) ^ 1, gbase + FF);
      // 32 loads of group g+1 just issued; async loads complete in order, so
      // ASYNCcnt <= 32 implies all of group g is resident in LDS.
      WAIT_ASYNC(32);
    } else {
      WAIT_ASYNC(0);             // drain for the final group
    }
    const int par = g & 1;
#pragma unroll
    for (int p = 0; p < GRP; ++p) {
      const float4 v4 =
          *reinterpret_cast<const float4*>(&sbuf[par][p][lane][0]);
      step(v4, gbase + p * (TT * FF));
    }
    gbase += FF;
  }
}

extern "C" void kernel_launch(void* const* d_in, const int* in_sizes, int n_in,
                              void* d_out, int out_size, void* d_ws, size_t ws_size,
                              hipStream_t stream) {
  (void)in_sizes; (void)n_in; (void)out_size; (void)d_ws; (void)ws_size;
  const float* mic_r = (const float*)d_in[0];
  const float* mic_i = (const float*)d_in[1];
  const float* ref_r = (const float*)d_in[2];
  const float* ref_i = (const float*)d_in[3];
  const float* c0r   = (const float*)d_in[4];
  const float* c0i   = (const float*)d_in[5];
  float* out = (float*)d_out;

  dim3 grid((FF + 31) / 32);   // 9 independent wave32 workgroups, one per WGP
  dim3 block(32);
  hipLaunchKernelGGL(aec_ipnlms_kernel, grid, block, 0, stream,
                     mic_r, mic_i, ref_r, ref_i, c0r, c0i, out);
}